// VoxelTransformer_26731876450583
// MI455X (gfx1250) — compile-verified
//
#include <hip/hip_runtime.h>
#include <cmath>

// Problem shape (fixed by setup_inputs in the reference)
#define B_ 4
#define N_ 8192
#define F_ 64
#define A_ 64
#define S_ 4096
#define K_ 3

typedef __attribute__((ext_vector_type(2))) float v2f;
typedef __attribute__((ext_vector_type(8))) float v8f;

// ---------------------------------------------------------------------------
// Kernel 1: farthest point sampling. One workgroup per batch; min-dist array
// lives in LDS; each step = distance update + workgroup argmax reduction.
// ---------------------------------------------------------------------------
#define FPS_THREADS 1024

__global__ void __launch_bounds__(FPS_THREADS)
fps_kernel(const float* __restrict__ spatial, int* __restrict__ sidx)
{
    __shared__ float s_dist[N_];                 // 32 KB
    __shared__ float s_redv[FPS_THREADS / 32];
    __shared__ int   s_redi[FPS_THREADS / 32];
    __shared__ int   s_last;

    const int b   = blockIdx.x;
    const int tid = threadIdx.x;
    const float* __restrict__ sp = spatial + (size_t)b * N_ * 5;

    for (int j = tid; j < N_; j += FPS_THREADS) s_dist[j] = __builtin_inff();
    if (tid == 0) {
        s_dist[0]    = -__builtin_inff();
        s_last       = 0;
        sidx[b * S_] = 0;
    }
    __syncthreads();

    for (int s = 1; s < S_; ++s) {
        const int last = s_last;
        const float p0 = sp[last * 5 + 0];
        const float p1 = sp[last * 5 + 1];
        const float p2 = sp[last * 5 + 2];
        const float p3 = sp[last * 5 + 3];
        const float p4 = sp[last * 5 + 4];

        float bestv = -__builtin_inff();
        int   besti = 0;
        for (int j = tid; j < N_; j += FPS_THREADS) {
            const float d0 = sp[j * 5 + 0] - p0;
            const float d1 = sp[j * 5 + 1] - p1;
            const float d2 = sp[j * 5 + 2] - p2;
            const float d3 = sp[j * 5 + 3] - p3;
            const float d4 = sp[j * 5 + 4] - p4;
            const float d  = d0 * d0 + d1 * d1 + d2 * d2 + d3 * d3 + d4 * d4;
            const float nd = fminf(s_dist[j], d);
            s_dist[j] = nd;
            if (nd > bestv) { bestv = nd; besti = j; }
        }
        // wave32 argmax reduction
        #pragma unroll
        for (int off = 16; off > 0; off >>= 1) {
            const float ov = __shfl_down(bestv, off, 32);
            const int   oi = __shfl_down(besti, off, 32);
            if (ov > bestv) { bestv = ov; besti = oi; }
        }
        const int wave = tid >> 5;
        if ((tid & 31) == 0) { s_redv[wave] = bestv; s_redi[wave] = besti; }
        __syncthreads();
        if (tid == 0) {
            float bv = s_redv[0];
            int   bi = s_redi[0];
            for (int w = 1; w < FPS_THREADS / 32; ++w)
                if (s_redv[w] > bv) { bv = s_redv[w]; bi = s_redi[w]; }
            s_dist[bi]       = -__builtin_inff();   // exclude from remaining
            s_last           = bi;
            sidx[b * S_ + s] = bi;
        }
        __syncthreads();
    }
}

// ---------------------------------------------------------------------------
// Kernel 2: kNN (k=3, self excluded) + neighbor-feature gather.
// One wave per sampled point; lane-local sorted top-3, shuffle merge.
// ---------------------------------------------------------------------------
__device__ __forceinline__ void top3_insert(float dd, int j, float d[3], int ix[3])
{
    if (dd < d[2]) {
        if (dd < d[0]) {
            d[2] = d[1]; ix[2] = ix[1];
            d[1] = d[0]; ix[1] = ix[0];
            d[0] = dd;   ix[0] = j;
        } else if (dd < d[1]) {
            d[2] = d[1]; ix[2] = ix[1];
            d[1] = dd;   ix[1] = j;
        } else {
            d[2] = dd;   ix[2] = j;
        }
    }
}

__global__ void __launch_bounds__(256)
knn_gather_kernel(const float* __restrict__ spatial, const float* __restrict__ x,
                  const int* __restrict__ sidx, float* __restrict__ neigh)
{
    const int lane = threadIdx.x & 31;
    const int wave = threadIdx.x >> 5;
    const int sid  = blockIdx.x * (blockDim.x >> 5) + wave;   // 0..B*S-1
    const int b    = sid / S_;
    const int s    = sid - b * S_;

    const int self = sidx[b * S_ + s];
    const float* __restrict__ sp = spatial + (size_t)b * N_ * 5;
    const float p0 = sp[self * 5 + 0];
    const float p1 = sp[self * 5 + 1];
    const float p2 = sp[self * 5 + 2];
    const float p3 = sp[self * 5 + 3];
    const float p4 = sp[self * 5 + 4];

    float d[3]  = { __builtin_inff(), __builtin_inff(), __builtin_inff() };
    int   ix[3] = { 0, 0, 0 };

    for (int j = lane; j < N_; j += 32) {
        if (j == self) continue;
        const float d0 = sp[j * 5 + 0] - p0;
        const float d1 = sp[j * 5 + 1] - p1;
        const float d2 = sp[j * 5 + 2] - p2;
        const float d3 = sp[j * 5 + 3] - p3;
        const float d4 = sp[j * 5 + 4] - p4;
        top3_insert(d0 * d0 + d1 * d1 + d2 * d2 + d3 * d3 + d4 * d4, j, d, ix);
    }

    // merge sorted triples across the wave
    #pragma unroll
    for (int off = 16; off > 0; off >>= 1) {
        float od[3]; int oi[3];
        #pragma unroll
        for (int t = 0; t < 3; ++t) {
            od[t] = __shfl_down(d[t], off, 32);
            oi[t] = __shfl_down(ix[t], off, 32);
        }
        float md[3]; int mi[3];
        int ai = 0, bi = 0;
        #pragma unroll
        for (int t = 0; t < 3; ++t) {
            if (d[ai] <= od[bi]) { md[t] = d[ai]; mi[t] = ix[ai]; ++ai; }
            else                 { md[t] = od[bi]; mi[t] = oi[bi]; ++bi; }
        }
        #pragma unroll
        for (int t = 0; t < 3; ++t) { d[t] = md[t]; ix[t] = mi[t]; }
    }

    const int n0 = __shfl(ix[0], 0, 32);
    const int n1 = __shfl(ix[1], 0, 32);
    const int n2 = __shfl(ix[2], 0, 32);

    const float* __restrict__ xb  = x + (size_t)b * N_ * F_;
    float* __restrict__       dst = neigh + (size_t)sid * K_ * F_;
    for (int f = lane; f < F_; f += 32) {
        dst[0 * F_ + f] = xb[(size_t)n0 * F_ + f];
        dst[1 * F_ + f] = xb[(size_t)n1 * F_ + f];
        dst[2 * F_ + f] = xb[(size_t)n2 * F_ + f];
    }
}

// ---------------------------------------------------------------------------
// Kernel 3: subtract-based vector attention on sampled features, built on
// V_WMMA_F32_16X16X4_F32 (exact f32, matches reference math).
// One wave owns a 16-row tile; five 16x64 LDS buffers per wave.
// ---------------------------------------------------------------------------
// dst[m][col] = bias[col] + sum_f src[m][f] * W[col][f]   (optionally ReLU)
__device__ __forceinline__ void wave_matmul(const float* __restrict__ W,
                                            const float* __restrict__ bias,
                                            const float* __restrict__ src,
                                            float* __restrict__ dst,
                                            int lane, bool do_relu)
{
    const int half = lane >> 4;   // lane half selects K {0,1} vs {2,3} and D rows
    const int lm   = lane & 15;
    for (int nn = 0; nn < 4; ++nn) {
        const int   col = nn * 16 + lm;
        const float bv  = bias[col];
        v8f acc = { bv, bv, bv, bv, bv, bv, bv, bv };
        #pragma unroll
        for (int kk = 0; kk < 16; ++kk) {
            const int k0 = kk * 4 + 2 * half;
            v2f a;  a.x  = src[lm * 64 + k0];  a.y  = src[lm * 64 + k0 + 1];
            v2f bm; bm.x = W[col * 64 + k0];   bm.y = W[col * 64 + k0 + 1];
            acc = __builtin_amdgcn_wmma_f32_16x16x4_f32(
                false, a, false, bm, (short)0, acc, false, false);
        }
        #pragma unroll
        for (int r = 0; r < 8; ++r) {
            float v = acc[r];
            if (do_relu) v = fmaxf(v, 0.0f);
            dst[(half * 8 + r) * 64 + col] = v;   // D: VGPR r -> row 8*half + r
        }
    }
}

// Final matmul: out[m][col] = bias[col] + sum_f src[m][f]*W[col][f] + resid[m][col]
__device__ __forceinline__ void wave_matmul_out(const float* __restrict__ W,
                                                const float* __restrict__ bias,
                                                const float* __restrict__ src,
                                                const float* __restrict__ resid,
                                                float* __restrict__ gout,
                                                int lane)
{
    const int half = lane >> 4;
    const int lm   = lane & 15;
    for (int nn = 0; nn < 4; ++nn) {
        const int   col = nn * 16 + lm;
        const float bv  = bias[col];
        v8f acc = { bv, bv, bv, bv, bv, bv, bv, bv };
        #pragma unroll
        for (int kk = 0; kk < 16; ++kk) {
            const int k0 = kk * 4 + 2 * half;
            v2f a;  a.x  = src[lm * 64 + k0];  a.y  = src[lm * 64 + k0 + 1];
            v2f bm; bm.x = W[col * 64 + k0];   bm.y = W[col * 64 + k0 + 1];
            acc = __builtin_amdgcn_wmma_f32_16x16x4_f32(
                false, a, false, bm, (short)0, acc, false, false);
        }
        #pragma unroll
        for (int r = 0; r < 8; ++r) {
            const int m = half * 8 + r;
            gout[(size_t)m * 64 + col] = acc[r] + resid[m * 64 + col];
        }
    }
}

#define ATTN_WAVES 2

__global__ void __launch_bounds__(ATTN_WAVES * 32)
attn_kernel(const float* __restrict__ x, const int* __restrict__ sidx,
            const float* __restrict__ w_in,  const float* __restrict__ b_in,
            const float* __restrict__ w_q,   const float* __restrict__ b_q,
            const float* __restrict__ w_k,   const float* __restrict__ b_k,
            const float* __restrict__ w_v,   const float* __restrict__ b_v,
            const float* __restrict__ w_h1,  const float* __restrict__ b_h1,
            const float* __restrict__ w_h2,  const float* __restrict__ b_h2,
            const float* __restrict__ w_out, const float* __restrict__ b_out,
            float* __restrict__ out)
{
    __shared__ float lds[ATTN_WAVES][5][16 * 64];   // 40 KB / block

    const int lane = threadIdx.x & 31;
    const int wave = threadIdx.x >> 5;
    const int tile = blockIdx.x * ATTN_WAVES + wave;      // 0..B*S/16-1
    const int tilesPerB = S_ / 16;
    const int b  = tile / tilesPerB;
    const int s0 = (tile - b * tilesPerB) * 16;

    float* xs = lds[wave][0];   // sampled features (kept for residual)
    float* tb = lds[wave][1];   // t
    float* bA = lds[wave][2];   // q -> (q-k) -> h -> softmax(w) -> w*v
    float* bB = lds[wave][3];   // k -> v
    float* bC = lds[wave][4];   // h1

    // Warm L2/L0 for weight matrices (global_prefetch_b8)
    __builtin_prefetch(w_in, 0, 2);
    __builtin_prefetch(w_q, 0, 2);
    __builtin_prefetch(w_k, 0, 2);
    __builtin_prefetch(w_v, 0, 2);
    __builtin_prefetch(w_h1, 0, 2);
    __builtin_prefetch(w_h2, 0, 2);
    __builtin_prefetch(w_out, 0, 2);

    // Gather xs tile via sampled indices
    const float* __restrict__ xb = x + (size_t)b * N_ * F_;
    const int* __restrict__   sb = sidx + b * S_ + s0;
    for (int e = lane; e < 16 * 64; e += 32) {
        const int m = e >> 6, f = e & 63;
        xs[e] = xb[(size_t)sb[m] * F_ + f];
    }
    __syncthreads();

    wave_matmul(w_in, b_in, xs, tb, lane, false);   // t
    __syncthreads();
    wave_matmul(w_q, b_q, tb, bA, lane, false);     // q
    __syncthreads();
    wave_matmul(w_k, b_k, tb, bB, lane, false);     // k
    __syncthreads();
    for (int e = lane; e < 16 * 64; e += 32) bA[e] -= bB[e];   // q - k
    __syncthreads();
    wave_matmul(w_v, b_v, tb, bB, lane, false);     // v
    __syncthreads();
    wave_matmul(w_h1, b_h1, bA, bC, lane, true);    // relu((q-k)W1 + b1)
    __syncthreads();
    wave_matmul(w_h2, b_h2, bC, bA, lane, false);   // h
    __syncthreads();

    // per-row softmax over the 64 channels (lanes 0..15 each own one row)
    if (lane < 16) {
        float mx = -__builtin_inff();
        for (int c = 0; c < 64; ++c) mx = fmaxf(mx, bA[lane * 64 + c]);
        float sum = 0.0f;
        for (int c = 0; c < 64; ++c) {
            const float e = __expf(bA[lane * 64 + c] - mx);
            bA[lane * 64 + c] = e;
            sum += e;
        }
        const float inv = 1.0f / sum;
        for (int c = 0; c < 64; ++c) bA[lane * 64 + c] *= inv;
    }
    __syncthreads();

    for (int e = lane; e < 16 * 64; e += 32) bA[e] *= bB[e];   // w * v
    __syncthreads();

    wave_matmul_out(w_out, b_out, bA, xs,
                    out + ((size_t)b * S_ + s0) * F_, lane);
}

// ---------------------------------------------------------------------------
// Host-side launcher
// ---------------------------------------------------------------------------
extern "C" void kernel_launch(void* const* d_in, const int* in_sizes, int n_in,
                              void* d_out, int out_size, void* d_ws, size_t ws_size,
                              hipStream_t stream)
{
    const float* x       = (const float*)d_in[0];
    const float* spatial = (const float*)d_in[1];
    const float* w_in    = (const float*)d_in[2];
    const float* b_in    = (const float*)d_in[3];
    const float* w_q     = (const float*)d_in[4];
    const float* b_q     = (const float*)d_in[5];
    const float* w_k     = (const float*)d_in[6];
    const float* b_k     = (const float*)d_in[7];
    const float* w_v     = (const float*)d_in[8];
    const float* b_v     = (const float*)d_in[9];
    const float* w_h1    = (const float*)d_in[10];
    const float* b_h1    = (const float*)d_in[11];
    const float* w_h2    = (const float*)d_in[12];
    const float* b_h2    = (const float*)d_in[13];
    const float* w_out   = (const float*)d_in[14];
    const float* b_out   = (const float*)d_in[15];

    float* out   = (float*)d_out;                       // [B,S,F]
    float* neigh = out + (size_t)B_ * S_ * F_;          // [B,S,K,F]
    int*   sidx  = (int*)d_ws;                          // [B,S]

    hipLaunchKernelGGL(fps_kernel, dim3(B_), dim3(FPS_THREADS), 0, stream,
                       spatial, sidx);

    hipLaunchKernelGGL(knn_gather_kernel, dim3((B_ * S_) / 8), dim3(256), 0, stream,
                       spatial, x, sidx, neigh);

    hipLaunchKernelGGL(attn_kernel, dim3((B_ * S_ / 16) / ATTN_WAVES),
                       dim3(ATTN_WAVES * 32), 0, stream,
                       x, sidx,
                       w_in, b_in, w_q, b_q, w_k, b_k, w_v, b_v,
                       w_h1, b_h1, w_h2, b_h2, w_out, b_out, out);
}